// SpeedPerturb_77163382440187
// MI455X (gfx1250) — compile-verified
//
#include <hip/hip_runtime.h>
#include <math.h>

// ---------------------------------------------------------------------------
// Kaldi LinearResample 16000 -> 14400 on MI455X (gfx1250).
//
// Roofline: ~0.77 GFLOP vs ~234 MB traffic -> bandwidth bound (~10us @ 23.3TB/s).
// Strategy: exact-fp32 GEMM formulation using V_WMMA_F32_16X16X4_F32.
//   out[b, 18k+i] = sum_{c=0..31} x[b, 20k-6+c] * EW[i][c]
// Tile = 16 k-blocks (M) x 32 phases (2 N-tiles of 16) x K=32 (8 wmma steps).
// One tile = 288 contiguous outputs. Input staged into LDS by the Tensor Data
// Mover (TENSOR_LOAD_TO_LDS, TENSORcnt) for interior tiles; edge tiles use a
// guarded scalar path. Output staged LDS then stored as b128.
// ---------------------------------------------------------------------------

typedef __attribute__((ext_vector_type(2))) float v2f;
typedef __attribute__((ext_vector_type(4))) float v4f;
typedef __attribute__((ext_vector_type(8))) float v8f;
typedef __attribute__((ext_vector_type(4))) unsigned int u32x4;
typedef __attribute__((ext_vector_type(8))) unsigned int u32x8;

#define ORIGF   16000
#define NEWFQ   14400
#define UPF     18
#define TIN     480000
#define TOUT    432000
#define BATCH   64
#define TILES_PER_ROW 1500              // 24000 k-blocks / 16 per tile
#define NTILES  (BATCH * TILES_PER_ROW) // 96000
#define EWS     34                      // EW row stride (bank-conflict-free, 8B-aligned v2f)
#define INSPAN  352                     // staged input floats per tile (need 332)
#define INSTG   360
#define OUTSTG  288
#define PI_D    3.14159265358979323846

// LDS layout (floats): [0,1088) EW[32][34]; per-wave in-stage @1088+wid*360;
// per-wave out-stage @3968+wid*288. Total 6272 floats = 25088 B.
// (Single __shared__ array -> LDS base offset 0; in-stage byte addr is
//  4*(1088 + wid*360), used in the TDM descriptor.)
#define LDS_FLOATS 6272

__global__ __launch_bounds__(256) void
SpeedPerturb_resample_wmma(const float* __restrict__ x, float* __restrict__ out)
{
    __shared__ float lds[LDS_FLOATS];
    const int tid  = threadIdx.x;
    const int lane = tid & 31;
    const int wid  = tid >> 5;
    const int half = lane >> 4;   // 0: lanes 0-15, 1: lanes 16-31
    const int l16  = lane & 15;

    // ---- Build expanded weight matrix EW[phase][col] in fp64 (matches ref) ----
    for (int p = tid; p < 32 * EWS; p += 256) lds[p] = 0.0f;
    __syncthreads();
    if (tid < UPF) {
        const double lc = 0.99 * 0.5 * (double)NEWFQ;       // lowpass cutoff = 7128
        const double ww = 6.0 / (2.0 * lc);                 // window width (s)
        const double ot = (double)tid / (double)NEWFQ;      // output_t
        const double fi = ceil((ot - ww) * (double)ORIGF);  // min_input_index
        const double mi = floor((ot + ww) * (double)ORIGF); // max_input_index
        const int n = (int)(mi - fi) + 1;
        for (int j = 0; j < n; ++j) {
            double dt = (fi + (double)j) / (double)ORIGF - ot;
            double w  = 0.0;
            if (fabs(dt) < ww) w = 0.5 * (1.0 + cos(2.0 * PI_D * lc / 6.0 * dt));
            if (dt != 0.0) w *= sin(2.0 * PI_D * lc * dt) / (PI_D * dt);
            else           w *= 2.0 * lc;
            w /= (double)ORIGF;
            int col = (int)fi + j + 6;        // cols land exactly in [0,32)
            if (col >= 0 && col < 32) lds[tid * EWS + col] = (float)w;
        }
    }
    __syncthreads();

    // ---- Preload tile-invariant B fragments: B[k,n] = EW[n][k] ----
    // Layout (mirrors documented 16x4 A / 16x16 C striping):
    //   VGPR0: lanes0-15 -> K=4s,  lanes16-31 -> K=4s+2
    //   VGPR1: lanes0-15 -> K=4s+1,lanes16-31 -> K=4s+3
    v2f bfrag[8][2];
#pragma unroll
    for (int s = 0; s < 8; ++s) {
#pragma unroll
        for (int t = 0; t < 2; ++t) {
            const int phase = l16 + 16 * t;            // phases >=18 are zero rows
            const int col   = 4 * s + 2 * half;
            bfrag[s][t] = *(const v2f*)&lds[phase * EWS + col];
        }
    }

    float* inS  = &lds[1088 + wid * INSTG];
    float* outS = &lds[3968 + wid * OUTSTG];
    const unsigned inS_ldsbyte =
        __builtin_amdgcn_readfirstlane((unsigned)((1088 + wid * INSTG) * 4));

    const int totalWaves = (gridDim.x * blockDim.x) >> 5;
    for (int tile = (blockIdx.x * blockDim.x + tid) >> 5; tile < NTILES; tile += totalWaves) {
        const int b  = tile / TILES_PER_ROW;
        const int tk = tile - b * TILES_PER_ROW;
        const int g0 = 320 * tk - 6;                   // first staged input col
        const float* xr = x + b * TIN;

        if (tk > 0 && tk < TILES_PER_ROW - 1) {
            // ---- Interior tile: Tensor Data Mover, 1-D tile of 352 dwords ----
            const unsigned long long ga = (unsigned long long)(const void*)(xr + g0);
            const unsigned ga_lo = __builtin_amdgcn_readfirstlane((unsigned)ga);
            const unsigned ga_hi = __builtin_amdgcn_readfirstlane((unsigned)(ga >> 32));
            const unsigned td0   = __builtin_amdgcn_readfirstlane((unsigned)(TIN - g0));

            u32x4 dg0;
            dg0.x = 1u;                                       // count=1, user D#
            dg0.y = inS_ldsbyte;                              // lds_addr (bytes)
            dg0.z = ga_lo;                                    // global_addr[31:0]
            dg0.w = (ga_hi & 0x01FFFFFFu) | (2u << 30);       // addr[56:32] | type=2

            u32x8 dg1;
            dg1.s0 = (2u << 16);                              // data_size = 4 bytes
            dg1.s1 = (td0 & 0xFFFFu) << 16;                   // tensor_dim0[15:0]
            dg1.s2 = ((td0 >> 16) & 0xFFFFu) | (1u << 16);    // tensor_dim0[31:16], tensor_dim1=1
            dg1.s3 = ((unsigned)INSPAN) << 16;                // tile_dim0 = 352
            dg1.s4 = 0u;                                      // tile_dim1/2 = 0 (unused)
            dg1.s5 = 0u;                                      // tensor_dim0_stride = 0
            dg1.s6 = 0u;
            dg1.s7 = 0u;                                      // tensor_dim1_stride = 0

            // VADDR2/VADDR3 omitted (NULL) -> <=2D tensor, 2 SGPR groups.
            asm volatile("tensor_load_to_lds %0, %1"
                         :: "s"(dg0), "s"(dg1) : "memory");
            __builtin_amdgcn_s_wait_tensorcnt(0);
        } else {
            // ---- Edge tile: guarded coalesced loads with zero-fill ----
            for (int j = lane; j < INSPAN; j += 32) {
                const int col = g0 + j;
                float v = 0.0f;
                if (col >= 0 && col < TIN) v = xr[col];
                inS[j] = v;
            }
            asm volatile("s_wait_dscnt 0" ::: "memory");
        }

        // ---- 8 K-steps, two 16-wide N-tiles: D = A x B + C (exact fp32) ----
        v8f acc0 = {};
        v8f acc1 = {};
#pragma unroll
        for (int s = 0; s < 8; ++s) {
            // A[m, k]: m = l16 (rows 20 floats apart), k = 4s + 2*half (+1)
            v2f a = *(const v2f*)&inS[20 * l16 + 4 * s + 2 * half];
            acc0 = __builtin_amdgcn_wmma_f32_16x16x4_f32(
                       false, a, false, bfrag[s][0], (short)0, acc0, false, false);
            acc1 = __builtin_amdgcn_wmma_f32_16x16x4_f32(
                       false, a, false, bfrag[s][1], (short)0, acc1, false, false);
        }

        // ---- Scatter D tiles into contiguous 288-float LDS stage ----
        // C/D layout: VGPR v, lanes0-15 -> M=v, lanes16-31 -> M=v+8; N=l16.
#pragma unroll
        for (int v = 0; v < 8; ++v) {
            const int m = v + 8 * half;
            outS[m * UPF + l16] = acc0[v];             // phases 0..15
        }
        if (l16 < 2) {
#pragma unroll
            for (int v = 0; v < 8; ++v) {
                const int m = v + 8 * half;
                outS[m * UPF + 16 + l16] = acc1[v];    // phases 16,17
            }
        }
        asm volatile("s_wait_dscnt 0" ::: "memory");

        // ---- Coalesced b128 stores: 288 contiguous outputs ----
        float* orow = out + b * TOUT + OUTSTG * tk;    // 16B-aligned
        for (int j4 = lane; j4 < OUTSTG / 4; j4 += 32) {
            v4f v = *(const v4f*)&outS[4 * j4];
            *(v4f*)&orow[4 * j4] = v;
        }
        // Ensure all DS reads of inS/outS completed before next tile's TDM/stores.
        asm volatile("s_wait_dscnt 0" ::: "memory");
    }
}

extern "C" void kernel_launch(void* const* d_in, const int* in_sizes, int n_in,
                              void* d_out, int out_size, void* d_ws, size_t ws_size,
                              hipStream_t stream)
{
    (void)in_sizes; (void)n_in; (void)out_size; (void)d_ws; (void)ws_size;
    const float* x = (const float*)d_in[0];
    float* out = (float*)d_out;
    // 1500 blocks * 8 waves = 12000 waves -> 8 tiles per wave, no tail divergence.
    hipLaunchKernelGGL(SpeedPerturb_resample_wmma, dim3(1500), dim3(256), 0, stream,
                       x, out);
}